// AtlasRayMarching_10153302688392
// MI455X (gfx1250) — compile-verified
//
#include <hip/hip_runtime.h>

typedef __attribute__((ext_vector_type(2))) float v2f;
typedef __attribute__((ext_vector_type(8))) float v8f;

// Problem constants (match the reference file).
#define NXc 160
#define NYc 160
#define NZc 64
#define Bc  4
#define Cc  32
#define Hc  120
#define Wc  160
constexpr int       HWc  = Hc * Wc;                       // 19200
constexpr long long NVPB = (long long)NXc * NYc * NZc;    // 1,638,400 voxels per batch
constexpr long long NV   = NVPB * Bc;                     // 6,553,600 voxels total

__global__ __launch_bounds__(256) void backproject_wmma_kernel(
    const float* __restrict__ origin,      // [B,3]
    const float* __restrict__ projection,  // [B,3,4]
    const float* __restrict__ features,    // [B,C,H,W]
    const float* __restrict__ voxel_size,  // [1]
    float* __restrict__ out)               // volume [B,C,NX,NY,NZ] then valid [B,1,NX,NY,NZ]
{
    const int lane = threadIdx.x & 31;
    const long long gid      = (long long)blockIdx.x * blockDim.x + threadIdx.x;
    const long long waveBase = gid & ~31LL;

    // Wave-uniform voxel decode: NZ=64 and waves are 32-aligned, so a wave
    // covers one contiguous half z-column of a single (b, ix, iy).
    const int       b     = (int)(waveBase / NVPB);
    const long long rBase = waveBase - (long long)b * NVPB;
    const int       ix    = (int)(rBase / ((long long)NYc * NZc));
    const int       iy    = (int)((rBase / NZc) % NYc);
    const int       izB   = (int)(rBase % NZc);            // 0 or 32

    const float vs = voxel_size[0];
    const float ox = origin[b * 3 + 0];
    const float oy = origin[b * 3 + 1];
    const float oz = origin[b * 3 + 2];
    const float wx = (float)ix * vs + ox;                  // constant across the wave
    const float wy = (float)iy * vs + oy;

    // ---- A operand: P (3x4) zero-padded to 16x4. Layout (32-bit A 16x4):
    // lanes 0-15 hold M=lane with K=0 (v0) / K=1 (v1); lanes 16-31 hold K=2 / K=3.
    const int m  = lane & 15;
    const int kp = (lane < 16) ? 0 : 2;
    v2f a; a.x = 0.0f; a.y = 0.0f;
    if (m < 3) {
        const float* P = projection + b * 12 + m * 4;
        a.x = P[kp];
        a.y = P[kp + 1];
    }

    // ---- B operands: columns = homogeneous world coords of 16 voxels.
    // B (4x16) layout mirrors A: lanes 0-15 rows K=0 (v0) / K=1 (v1),
    // lanes 16-31 rows K=2 (v0) / K=3 (v1); lane&15 selects column N.
    const int n = lane & 15;
    v2f b1, b2;
    if (lane < 16) {
        b1.x = wx;  b1.y = wy;
        b2.x = wx;  b2.y = wy;
    } else {
        b1.x = (float)(izB + n)      * vs + oz;  b1.y = 1.0f;   // wz rows
        b2.x = (float)(izB + 16 + n) * vs + oz;  b2.y = 1.0f;
    }

    // D = A*B + 0:  D[i,j] = cam_i of voxel j (i<3), zeros elsewhere.
    v8f czero = {0.f, 0.f, 0.f, 0.f, 0.f, 0.f, 0.f, 0.f};
    v8f d1 = __builtin_amdgcn_wmma_f32_16x16x4_f32(false, a, false, b1, (short)0, czero, false, false);
    v8f d2 = __builtin_amdgcn_wmma_f32_16x16x4_f32(false, a, false, b2, (short)0, czero, false, false);

    // Lane j (j<16): d1[0..2] = cam x/y/z of voxel z=izB+j; d2 covers izB+16+j.
    // Move tile-2 results into lanes 16..31 so every lane owns its own voxel.
    const float cx2 = __shfl(d2[0], lane & 15, 32);
    const float cy2 = __shfl(d2[1], lane & 15, 32);
    const float cz2 = __shfl(d2[2], lane & 15, 32);
    const float camx = (lane < 16) ? d1[0] : cx2;
    const float camy = (lane < 16) ? d1[1] : cy2;
    const float camz = (lane < 16) ? d1[2] : cz2;

    // Pixel coords: IEEE divide + round-half-even to match jnp.round exactly.
    const float pz = camz;
    const int px = (int)rintf(camx / pz);
    const int py = (int)rintf(camy / pz);
    const bool valid = (px >= 0) && (py >= 0) && (px < Wc) && (py < Hc) && (pz > 0.0f);

    const int pyc  = py < 0 ? 0 : (py > Hc - 1 ? Hc - 1 : py);
    const int pxc  = px < 0 ? 0 : (px > Wc - 1 ? Wc - 1 : px);
    const int flat = pyc * Wc + pxc;
    const float vmask = valid ? 1.0f : 0.0f;

    // Validity mask output (streamed, never re-read -> non-temporal).
    __builtin_nontemporal_store(vmask, out + (long long)Bc * Cc * NVPB + gid);

    // Channel loop: gather (L2-resident features) and stream out the volume.
    // Lanes write 32 consecutive dwords per channel -> one full cacheline/wave.
    const long long r  = rBase + (long long)lane;
    const float* __restrict__ fp = features + (long long)b * Cc * HWc + flat;
    float*       __restrict__ op = out      + (long long)b * Cc * NVPB + r;
#pragma unroll
    for (int c = 0; c < Cc; ++c) {
        const float v = fp[(long long)c * HWc] * vmask;   // clamped addr: always safe
        __builtin_nontemporal_store(v, op + (long long)c * NVPB);
    }
}

extern "C" void kernel_launch(void* const* d_in, const int* in_sizes, int n_in,
                              void* d_out, int out_size, void* d_ws, size_t ws_size,
                              hipStream_t stream) {
    const float* origin     = (const float*)d_in[0];
    const float* projection = (const float*)d_in[1];
    const float* features   = (const float*)d_in[2];
    const float* voxel_size = (const float*)d_in[3];
    float*       out        = (float*)d_out;

    // 6,553,600 voxels -> 25,600 blocks of 256 (8 waves), no tail (EXEC all-1s
    // at the WMMAs, as the ISA requires).
    dim3 block(256);
    dim3 grid((unsigned)(NV / 256));
    backproject_wmma_kernel<<<grid, block, 0, stream>>>(origin, projection, features,
                                                        voxel_size, out);
}